// RMAC_41721312314087
// MI455X (gfx1250) — compile-verified
//
#include <hip/hip_runtime.h>
#include <cstdint>
#include <cstddef>

// ---------------------------------------------------------------------------
// RMAC max-pooling for x:(64,37,37,512) f32, 14 static regions, out:(64,14,512)
// Memory-bound: exercise CDNA5 async global->LDS DMA (ASYNCcnt) pipeline.
// ---------------------------------------------------------------------------

#define B_   64
#define H_   37
#define W_   37
#define C_   512
#define NREG 14
#define NITV 6
#define PIX_BYTES 2048u          // 512 f32 per pixel
#define RING 16                  // LDS ring slots
#define DEPTH 8                  // outstanding async copies per wave

// Region tables (match reference rmac_regions(37,37,3) enumeration order).
// x-intervals: 0:(0,37) 1:(0,24) 2:(13,24) 3:(0,18) 4:(9,18) 5:(19,18)
__constant__ int c_r_itv[NREG] = {0, 1, 2, 1, 2, 3, 4, 5, 3, 4, 5, 3, 4, 5};
__constant__ int c_r_oy [NREG] = {0, 0, 0,13,13, 0, 0, 0, 9, 9, 9,19,19,19};
__constant__ int c_r_th [NREG] = {37,24,24,24,24,18,18,18,18,18,18,18,18,18};
__constant__ int c_r_ox [NREG] = {0, 0,13, 0,13, 0, 9,19, 0, 9,19, 0, 9,19};

__device__ __forceinline__ float4 f4max(float4 a, float4 b) {
    return make_float4(fmaxf(a.x, b.x), fmaxf(a.y, b.y),
                       fmaxf(a.z, b.z), fmaxf(a.w, b.w));
}

// Issue one per-lane 16B async copy: global (gaddr) -> LDS (dst byte offset).
__device__ __forceinline__ void async_cp16(uint32_t lds_dst, uint64_t gaddr) {
    asm volatile("global_load_async_to_lds_b128 %0, %1, off"
                 :: "v"(lds_dst), "v"(gaddr) : "memory");
}
__device__ __forceinline__ void wait_async_le7() {
    asm volatile("s_wait_asynccnt 0x7" ::: "memory");
}
__device__ __forceinline__ void wait_async_0() {
    asm volatile("s_wait_asynccnt 0x0" ::: "memory");
}

// ---------------------------------------------------------------------------
// Kernel 1: per (batch,row): stream 37 pixels through async-LDS ring, keep
// running max for the 6 x-intervals. rowmax layout: [b][y][itv][c].
// ---------------------------------------------------------------------------
__global__ __launch_bounds__(128) void rmac_rowmax_kernel(
    const float* __restrict__ in, float* __restrict__ rowmax) {
    const int b   = blockIdx.x / H_;
    const int y   = blockIdx.x % H_;
    const int tid = threadIdx.x;          // 128 threads -> 4 channels each

    __shared__ alignas(16) float buf[RING * C_];   // 32 KB ring
    const uint32_t lds0 =
        (uint32_t)(uintptr_t)&buf[0] + (uint32_t)tid * 16u;

    uint64_t gi = (uint64_t)(uintptr_t)in
                + (uint64_t)((b * H_ + y) * W_) * (uint64_t)PIX_BYTES
                + (uint64_t)tid * 16u;

    // Prologue: fill the pipeline (row has 37 px >= DEPTH).
    #pragma unroll
    for (int i = 0; i < DEPTH; ++i) {
        async_cp16(lds0 + (uint32_t)(i & (RING - 1)) * PIX_BYTES, gi);
        gi += PIX_BYTES;
    }

    const float NEG = -__FLT_MAX__;
    float4 m0 = make_float4(NEG, NEG, NEG, NEG);
    float4 m1 = m0, m2 = m0, m3 = m0, m4 = m0, m5 = m0;

    // Main loop: consume pixel p, issue pixel p+DEPTH.
    for (int p = 0; p < W_ - DEPTH; ++p) {
        wait_async_le7();
        const float4 v =
            *(const float4*)(&buf[(p & (RING - 1)) * C_ + tid * 4]);
        m0 = f4max(m0, v);                       // (0,37)
        if (p < 24)            m1 = f4max(m1, v); // (0,24)
        if (p >= 13)           m2 = f4max(m2, v); // (13,24)
        if (p < 18)            m3 = f4max(m3, v); // (0,18)
        if (p >= 9 && p < 27)  m4 = f4max(m4, v); // (9,18)
        if (p >= 19)           m5 = f4max(m5, v); // (19,18)
        async_cp16(lds0 + (uint32_t)((p + DEPTH) & (RING - 1)) * PIX_BYTES, gi);
        gi += PIX_BYTES;
    }
    // Drain.
    wait_async_0();
    for (int p = W_ - DEPTH; p < W_; ++p) {
        const float4 v =
            *(const float4*)(&buf[(p & (RING - 1)) * C_ + tid * 4]);
        m0 = f4max(m0, v);
        if (p < 24)            m1 = f4max(m1, v);
        if (p >= 13)           m2 = f4max(m2, v);
        if (p < 18)            m3 = f4max(m3, v);
        if (p >= 9 && p < 27)  m4 = f4max(m4, v);
        if (p >= 19)           m5 = f4max(m5, v);
    }

    float* o = rowmax + ((size_t)(b * H_ + y) * NITV) * C_ + tid * 4;
    *(float4*)(o + 0 * C_) = m0;
    *(float4*)(o + 1 * C_) = m1;
    *(float4*)(o + 2 * C_) = m2;
    *(float4*)(o + 3 * C_) = m3;
    *(float4*)(o + 4 * C_) = m4;
    *(float4*)(o + 5 * C_) = m5;
}

// ---------------------------------------------------------------------------
// Kernel 2: per (batch,region): max-reduce rowmax over the region's rows.
// ---------------------------------------------------------------------------
__global__ __launch_bounds__(128) void rmac_reduce_kernel(
    const float* __restrict__ rowmax, float* __restrict__ out) {
    const int b   = blockIdx.x / NREG;
    const int r   = blockIdx.x % NREG;
    const int tid = threadIdx.x;

    const int itv = c_r_itv[r];
    const int oy  = c_r_oy[r];
    const int th  = c_r_th[r];

    const float* p = rowmax
        + (((size_t)(b * H_ + oy) * NITV + itv) * C_) + tid * 4;

    const float NEG = -__FLT_MAX__;
    float4 m = make_float4(NEG, NEG, NEG, NEG);
    for (int yy = 0; yy < th; ++yy) {
        m = f4max(m, *(const float4*)p);
        p += (size_t)NITV * C_;
    }
    *(float4*)(out + ((size_t)(b * NREG + r)) * C_ + tid * 4) = m;
}

// ---------------------------------------------------------------------------
// Fallback (workspace too small): per (batch,region) direct streaming with
// the same async-LDS ring. Input fits MI455X's 192MB L2, so the ~4.8x
// re-reads mostly hit on-chip.
// ---------------------------------------------------------------------------
__global__ __launch_bounds__(128) void rmac_direct_kernel(
    const float* __restrict__ in, float* __restrict__ out) {
    const int b   = blockIdx.x / NREG;
    const int r   = blockIdx.x % NREG;
    const int tid = threadIdx.x;

    const int ox = c_r_ox[r];
    const int oy = c_r_oy[r];
    const int w  = c_r_th[r];              // square regions: th == tw

    __shared__ alignas(16) float buf[RING * C_];
    const uint32_t lds0 =
        (uint32_t)(uintptr_t)&buf[0] + (uint32_t)tid * 16u;

    uint64_t gi = (uint64_t)(uintptr_t)in
                + (uint64_t)((b * H_ + oy) * W_ + ox) * (uint64_t)PIX_BYTES
                + (uint64_t)tid * 16u;
    const uint64_t rowskip = (uint64_t)(W_ - w) * (uint64_t)PIX_BYTES;

    int ix = 0, cnt = 0;
    const int total = w * w;               // >= 324 > DEPTH always

    #pragma unroll
    for (int i = 0; i < DEPTH; ++i) {
        async_cp16(lds0 + (uint32_t)(cnt & (RING - 1)) * PIX_BYTES, gi);
        ++cnt; gi += PIX_BYTES;
        if (++ix == w) { ix = 0; gi += rowskip; }
    }

    const float NEG = -__FLT_MAX__;
    float4 m = make_float4(NEG, NEG, NEG, NEG);

    for (int p = 0; p < total - DEPTH; ++p) {
        wait_async_le7();
        m = f4max(m, *(const float4*)(&buf[(p & (RING - 1)) * C_ + tid * 4]));
        async_cp16(lds0 + (uint32_t)(cnt & (RING - 1)) * PIX_BYTES, gi);
        ++cnt; gi += PIX_BYTES;
        if (++ix == w) { ix = 0; gi += rowskip; }
    }
    wait_async_0();
    for (int p = total - DEPTH; p < total; ++p) {
        m = f4max(m, *(const float4*)(&buf[(p & (RING - 1)) * C_ + tid * 4]));
    }

    *(float4*)(out + ((size_t)(b * NREG + r)) * C_ + tid * 4) = m;
}

// ---------------------------------------------------------------------------
extern "C" void kernel_launch(void* const* d_in, const int* in_sizes, int n_in,
                              void* d_out, int out_size, void* d_ws,
                              size_t ws_size, hipStream_t stream) {
    (void)in_sizes; (void)n_in; (void)out_size;
    const float* x  = (const float*)d_in[0];
    float* out      = (float*)d_out;

    const size_t need = (size_t)B_ * H_ * NITV * C_ * sizeof(float); // 27.8 MB
    if (ws_size >= need && d_ws != nullptr) {
        float* rowmax = (float*)d_ws;
        rmac_rowmax_kernel<<<B_ * H_,   128, 0, stream>>>(x, rowmax);
        rmac_reduce_kernel<<<B_ * NREG, 128, 0, stream>>>(rowmax, out);
    } else {
        rmac_direct_kernel<<<B_ * NREG, 128, 0, stream>>>(x, out);
    }
}